// SpaceTimeRegionalConv_1580547968787
// MI455X (gfx1250) — compile-verified
//
#include <hip/hip_runtime.h>
#include <hip/hip_bf16.h>

// ---------------------------------------------------------------------------
// SpaceTimeRegionalConv on MI455X (gfx1250, wave32, WMMA)
//   y = conv_t(x, W, KT=3) + b           (per-(b,a) GEMM, M=L, N=C, K=C*KT)
//   out = relu((sum_a y + sum_l y - y)/sqrt(L+A-2) + x)
// fp32 end to end via V_WMMA_F32_16X16X4_F32 (exact reference numerics).
// Each wave computes a 2x2 grid of 16x16 tiles: 4 independent WMMA chains,
// B-fragments shared across the two A-fragments (8 LDS dwords / 4 WMMAs).
// ---------------------------------------------------------------------------

constexpr int Bn = 32;    // batch
constexpr int Ln = 128;   // length (time)
constexpr int An = 64;    // agents
constexpr int Cn = 128;   // channels (in == out)
constexpr int KTn = 3;    // temporal kernel

typedef float v2f __attribute__((ext_vector_type(2)));
typedef float v4f __attribute__((ext_vector_type(4)));
typedef float v8f __attribute__((ext_vector_type(8)));

// --- K1 LDS layout (bank-conflict-free fragment reads) ----------------------
constexpr int XS_STRIDE = 132;                   // padded row stride (x tile)
constexpr int XS_ROWS   = Ln + 2;                // halo rows l = -1 .. 128
constexpr int XS_FLOATS = XS_ROWS * XS_STRIDE;   // 17160
constexpr int WS_CI_STRIDE = 136;                // padded co-row stride (W)
constexpr int WS_KT_STRIDE = Cn * WS_CI_STRIDE;  // 17408
constexpr int WS_FLOATS    = KTn * WS_KT_STRIDE; // 52224
constexpr int K1_LDS_BYTES = (XS_FLOATS + WS_FLOATS) * 4;  // 277536 < 320KB

// ---------------------------------------------------------------------------
// K1: per-(b,a) temporal conv as WMMA GEMM.  8 waves; wave w owns l-tiles
// {2*(w&3), 2*(w&3)+1} and c_out half (w>>2); inner loop over 2 n-tile pairs.
// ---------------------------------------------------------------------------
__global__ void __launch_bounds__(256, 1)
stconv_wmma(const float* __restrict__ x, const float* __restrict__ W,
            const float* __restrict__ bias, float* __restrict__ y)
{
    extern __shared__ float lds[];
    float* xs  = lds;              // [130][132]  x tile with zero halo
    float* wsm = lds + XS_FLOATS;  // [3][128 ci][136] = W[kt][ci][co]

    const int ba  = blockIdx.x;        // 0 .. B*A-1
    const int b   = ba >> 6;
    const int a   = ba & (An - 1);
    const int tid = threadIdx.x;

    const float* xba = x + ((long)b * Ln * An + a) * Cn;  // + l*(An*Cn) + c

    // stage x tile (coalesced global reads, conflict-free LDS writes)
    for (int i = tid; i < XS_ROWS * Cn; i += 256) {
        int r = i >> 7;            // 0..129
        int c = i & (Cn - 1);
        int l = r - 1;
        float v = 0.0f;
        if (l >= 0 && l < Ln) v = xba[(long)l * (An * Cn) + c];
        xs[r * XS_STRIDE + c] = v;
    }
    // stage W: global-coalesced read of [co][ci][kt], scatter to [kt][ci][co]
    for (int i = tid; i < Cn * Cn * KTn; i += 256) {
        int co = i / (Cn * KTn);
        int r  = i - co * (Cn * KTn);
        int ci = r / KTn;
        int kt = r - ci * KTn;
        wsm[kt * WS_KT_STRIDE + ci * WS_CI_STRIDE + co] = W[i];
    }
    __syncthreads();

    const int wave = tid >> 5;
    const int lane = tid & 31;
    const int m    = lane & 15;    // A-row / B-col / D-col index
    const int grp  = lane >> 4;    // lane group: K offset 2*grp, D-row +8*grp
    const int lt   = (wave & 3) << 5;   // first l-tile origin: 0,32,64,96
    const int nh   = (wave >> 2) << 6;  // c_out half: 0 or 64

    float* ybase = y + (long)ba * Ln * Cn;   // y[b][a][l][c]

    for (int np = 0; np < 2; ++np) {
        const int n0 = nh + np * 32;         // two n-tiles: n0 and n0+16
        v8f acc00 = {0.f, 0.f, 0.f, 0.f, 0.f, 0.f, 0.f, 0.f};
        v8f acc01 = {0.f, 0.f, 0.f, 0.f, 0.f, 0.f, 0.f, 0.f};
        v8f acc10 = {0.f, 0.f, 0.f, 0.f, 0.f, 0.f, 0.f, 0.f};
        v8f acc11 = {0.f, 0.f, 0.f, 0.f, 0.f, 0.f, 0.f, 0.f};

        for (int kt = 0; kt < KTn; ++kt) {
            // A(l, ci): output row lt+m reads input lt+m+kt-1 -> xs row lt+m+kt
            const float* arow0 = xs + (lt + m + kt) * XS_STRIDE + 2 * grp;
            const float* arow1 = arow0 + 16 * XS_STRIDE;     // second l-tile
            // B(ci, co): rows ci0+2*grp, ci0+2*grp+1 ; col n0(+16)+m
            const float* wrow = wsm + kt * WS_KT_STRIDE
                                    + (2 * grp) * WS_CI_STRIDE + m;
            #pragma unroll 4
            for (int ci0 = 0; ci0 < Cn; ci0 += 4) {
                v2f a0 = *(const v2f*)(arow0 + ci0);         // 8B-aligned b64
                v2f a1 = *(const v2f*)(arow1 + ci0);
                const float* wp = wrow + ci0 * WS_CI_STRIDE;
                v2f b0, b1;
                b0.x = wp[n0];
                b0.y = wp[WS_CI_STRIDE + n0];
                b1.x = wp[n0 + 16];
                b1.y = wp[WS_CI_STRIDE + n0 + 16];
                acc00 = __builtin_amdgcn_wmma_f32_16x16x4_f32(
                            false, a0, false, b0, (short)0, acc00, false, false);
                acc01 = __builtin_amdgcn_wmma_f32_16x16x4_f32(
                            false, a0, false, b1, (short)0, acc01, false, false);
                acc10 = __builtin_amdgcn_wmma_f32_16x16x4_f32(
                            false, a1, false, b0, (short)0, acc10, false, false);
                acc11 = __builtin_amdgcn_wmma_f32_16x16x4_f32(
                            false, a1, false, b1, (short)0, acc11, false, false);
            }
        }

        const float bb0 = bias[n0 + m];
        const float bb1 = bias[n0 + 16 + m];
        #pragma unroll
        for (int v = 0; v < 8; ++v) {
            // D layout: lane -> col n0+m ; vgpr v + 8*grp -> row (l offset)
            int l0 = lt + v + (grp << 3);
            int l1 = l0 + 16;
            ybase[(long)l0 * Cn + n0 + m]      = acc00[v] + bb0;
            ybase[(long)l0 * Cn + n0 + 16 + m] = acc01[v] + bb1;
            ybase[(long)l1 * Cn + n0 + m]      = acc10[v] + bb0;
            ybase[(long)l1 * Cn + n0 + 16 + m] = acc11[v] + bb1;
        }
    }
}

// ---------------------------------------------------------------------------
// K2a: agent_sum[b][l][c] = sum_a y[b][a][l][c]   (deterministic order)
// ---------------------------------------------------------------------------
__global__ void __launch_bounds__(256)
agent_reduce(const float* __restrict__ y, float* __restrict__ agent_sum)
{
    int t  = blockIdx.x * 256 + threadIdx.x;   // 0 .. B*L*C/4
    int c4 = t & (Cn / 4 - 1);                 // 0..31
    int l  = (t >> 5) & (Ln - 1);
    int b  = t >> 12;
    const v4f* yp = (const v4f*)y + ((long)b * An * Ln + l) * (Cn / 4) + c4;
    v4f s = {0.f, 0.f, 0.f, 0.f};
    for (int a = 0; a < An; ++a)
        s += yp[(long)a * Ln * (Cn / 4)];
    ((v4f*)agent_sum)[((long)b * Ln + l) * (Cn / 4) + c4] = s;
}

// ---------------------------------------------------------------------------
// K2b: time_sum[b][a][c] = sum_l y[b][a][l][c]    (deterministic order)
// ---------------------------------------------------------------------------
__global__ void __launch_bounds__(256)
time_reduce(const float* __restrict__ y, float* __restrict__ time_sum)
{
    int t  = blockIdx.x * 256 + threadIdx.x;   // 0 .. B*A*C/4
    int c4 = t & (Cn / 4 - 1);
    int a  = (t >> 5) & (An - 1);
    int b  = t >> 11;
    const v4f* yp = (const v4f*)y + ((long)(b * An + a) * Ln) * (Cn / 4) + c4;
    v4f s = {0.f, 0.f, 0.f, 0.f};
    for (int l = 0; l < Ln; ++l)
        s += yp[l * (Cn / 4)];
    ((v4f*)time_sum)[(long)(b * An + a) * (Cn / 4) + c4] = s;
}

// ---------------------------------------------------------------------------
// K3: out[b][l][a][c] = relu((as + ts - y) / sqrt(L+A-2) + x)
// ---------------------------------------------------------------------------
__global__ void __launch_bounds__(256)
fuse_out(const float* __restrict__ x, const float* __restrict__ y,
         const float* __restrict__ agent_sum, const float* __restrict__ time_sum,
         float* __restrict__ out)
{
    long t = (long)blockIdx.x * 256 + threadIdx.x;  // float4 index over [B,L,A,C]
    int c4 = (int)(t & (Cn / 4 - 1));
    int a  = (int)(t >> 5) & (An - 1);
    int l  = (int)(t >> 11) & (Ln - 1);
    int b  = (int)(t >> 18);
    const float inv_scale = 0.07254762501100116f;   // 1/sqrt(190)

    v4f yv = ((const v4f*)y)[((long)(b * An + a) * Ln + l) * (Cn / 4) + c4];
    v4f as = ((const v4f*)agent_sum)[((long)b * Ln + l) * (Cn / 4) + c4];
    v4f ts = ((const v4f*)time_sum)[(long)(b * An + a) * (Cn / 4) + c4];
    v4f xv = ((const v4f*)x)[t];

    v4f g;
    #pragma unroll
    for (int i = 0; i < 4; ++i)
        g[i] = fmaxf((as[i] + ts[i] - yv[i]) * inv_scale + xv[i], 0.0f);
    ((v4f*)out)[t] = g;
}

// ---------------------------------------------------------------------------
extern "C" void kernel_launch(void* const* d_in, const int* in_sizes, int n_in,
                              void* d_out, int out_size, void* d_ws, size_t ws_size,
                              hipStream_t stream)
{
    const float* x    = (const float*)d_in[0];
    const float* W    = (const float*)d_in[1];
    const float* bias = (const float*)d_in[2];
    float* out = (float*)d_out;

    // workspace: y [B][A][L][C] | agent_sum [B][L][C] | time_sum [B][A][C]
    float* y         = (float*)d_ws;
    float* agent_sum = y + (long)Bn * An * Ln * Cn;
    float* time_sum  = agent_sum + (long)Bn * Ln * Cn;

    hipFuncSetAttribute((const void*)stconv_wmma,
                        hipFuncAttributeMaxDynamicSharedMemorySize, K1_LDS_BYTES);

    stconv_wmma<<<Bn * An, 256, K1_LDS_BYTES, stream>>>(x, W, bias, y);
    agent_reduce<<<(Bn * Ln * Cn / 4) / 256, 256, 0, stream>>>(y, agent_sum);
    time_reduce<<<(Bn * An * Cn / 4) / 256, 256, 0, stream>>>(y, time_sum);
    fuse_out<<<(Bn * Ln * An * Cn / 4) / 256, 256, 0, stream>>>(x, y, agent_sum,
                                                                time_sum, out);
}